// RoIAlignAda_31464930410636
// MI455X (gfx1250) — compile-verified
//
#include <hip/hip_runtime.h>

// RoIAlign (N=1024, C=256, 7x7, sample 2x2, bilinear) for MI455X / gfx1250.
//
// Strategy: memory-bound gather; pooling 196 samples -> 49 bins is cast as
//   Out(49x16) = P(49x196) x S(196x16)   per (roi, 16-channel tile)
// executed as 49 K-steps of V_WMMA_F32_16X16X4_F32 per wave (wave32).
// Each lane gathers 2 bilinear samples per K-step (8 global f32 loads),
// forming the B fragment; the constant pooling matrix P (0.25 entries)
// is synthesized per K-step as the A fragment. Accumulators = 4 M-tiles
// of v8f covering bins 0..63 (49 valid).

typedef __attribute__((ext_vector_type(2))) float v2f;
typedef __attribute__((ext_vector_type(8))) float v8f;

#define OUT_HW 7
#define SN     2
#define SGRID  14            // OUT_HW * SN
#define C_     256
#define H_     200
#define W_     272
#define NROI   1024
#define SCALE  0.25f

__global__ __launch_bounds__(256) void roi_align_wmma_kernel(
    const float* __restrict__ feats,   // (4, 256, 200, 272)
    const float* __restrict__ rois,    // (1024, 5)
    float* __restrict__ out)           // (1024, 256, 7, 7)
{
    __shared__ float4 sXg[SGRID];      // (x0, lx, valid, -)
    __shared__ float4 sYg[SGRID];      // (y0, ly, valid, -)

    const int roi = blockIdx.x >> 1;
    const int t   = threadIdx.x;

    // ---- per-roi sample geometry (shared by all 256 channels) ----
    if (t < 2 * SGRID) {
        const int axis = (t >= SGRID) ? 1 : 0;          // 0 = x, 1 = y
        const int i    = t - axis * SGRID;
        const float lo  = rois[roi * 5 + 1 + axis] * SCALE;
        const float hi  = rois[roi * 5 + 3 + axis] * SCALE;
        const float ext = fmaxf(hi - lo, 1.0f);
        const float bin = ext * (1.0f / (float)OUT_HW);
        const float p   = ((float)i + 0.5f) * (1.0f / (float)SN);
        const float crd = lo + p * bin;
        const float lim = axis ? (float)H_ : (float)W_;
        const float vld = (crd > -1.0f && crd < lim) ? 1.0f : 0.0f;
        float c  = fminf(fmaxf(crd, 0.0f), lim - 1.0f);
        float c0 = fminf(floorf(c), lim - 2.0f);
        float l  = c - c0;
        float4 g = make_float4(c0, l, vld, 0.0f);
        if (axis) sYg[i] = g; else sXg[i] = g;
    }
    __syncthreads();

    // ---- wave -> (roi, 16-channel tile) ----
    const int wave  = t >> 5;
    const int lane  = t & 31;
    const int half  = lane >> 4;       // lane half selects K pair (K = v + 2*half)
    const int ncol  = lane & 15;       // B/C/D column == channel within tile
    const int ctile = ((blockIdx.x & 1) << 3) + wave;   // 0..15
    const int c     = (ctile << 4) + ncol;
    const int b     = (int)rois[roi * 5];
    const float* __restrict__ plane =
        feats + (size_t)(b * C_ + c) * (size_t)(H_ * W_);

    v8f acc0 = {}, acc1 = {}, acc2 = {}, acc3 = {};

    // K = 196 samples, 4 per WMMA step -> exactly 49 steps, no tail.
    for (int k4 = 0; k4 < 49; ++k4) {
        const int sbase = (k4 << 2) + (half << 1);
        float val[2];
        int   bin[2];
        #pragma unroll
        for (int j = 0; j < 2; ++j) {
            const int s  = sbase + j;            // flattened 14x14 sample id
            const int sy = s / SGRID;
            const int sx = s - sy * SGRID;
            const float4 yg = sYg[sy];
            const float4 xg = sXg[sx];
            const int   y0 = (int)yg.x;
            const float ly = yg.y, hy = 1.0f - ly;
            const int   x0 = (int)xg.x;
            const float lx = xg.y, hx = 1.0f - lx;
            const float* p = plane + y0 * W_ + x0;
            const float v00 = p[0];
            const float v01 = p[1];
            const float v10 = p[W_];
            const float v11 = p[W_ + 1];
            const float v = hy * (hx * v00 + lx * v01)
                          + ly * (hx * v10 + lx * v11);
            val[j] = v * (yg.z * xg.z);          // zero out invalid samples
            bin[j] = (sy >> 1) * OUT_HW + (sx >> 1);   // pooling target bin
        }

        // B fragment (4x16 f32): VGPR v, lane-half h -> K = v + 2h
        v2f bf; bf[0] = val[0]; bf[1] = val[1];

        // A fragment (16x4 f32): lane M = ncol, VGPR v -> K = v + 2*half.
        // P[bin][s] = 0.25 iff this sample pools into row `bin`.
        {
            v2f af;
            af[0] = (bin[0] == ncol)        ? 0.25f : 0.0f;
            af[1] = (bin[1] == ncol)        ? 0.25f : 0.0f;
            acc0 = __builtin_amdgcn_wmma_f32_16x16x4_f32(
                false, af, false, bf, (short)0, acc0, false, false);
        }
        {
            v2f af;
            af[0] = (bin[0] == 16 + ncol)   ? 0.25f : 0.0f;
            af[1] = (bin[1] == 16 + ncol)   ? 0.25f : 0.0f;
            acc1 = __builtin_amdgcn_wmma_f32_16x16x4_f32(
                false, af, false, bf, (short)0, acc1, false, false);
        }
        {
            v2f af;
            af[0] = (bin[0] == 32 + ncol)   ? 0.25f : 0.0f;
            af[1] = (bin[1] == 32 + ncol)   ? 0.25f : 0.0f;
            acc2 = __builtin_amdgcn_wmma_f32_16x16x4_f32(
                false, af, false, bf, (short)0, acc2, false, false);
        }
        {
            v2f af;
            af[0] = (bin[0] == 48 + ncol)   ? 0.25f : 0.0f;
            af[1] = (bin[1] == 48 + ncol)   ? 0.25f : 0.0f;
            acc3 = __builtin_amdgcn_wmma_f32_16x16x4_f32(
                false, af, false, bf, (short)0, acc3, false, false);
        }
    }

    // ---- store: D layout lane L, VGPR r -> M = r + 8*half, N = L&15 ----
    float* obase = out + ((size_t)roi * C_ + c) * (OUT_HW * OUT_HW);
    #pragma unroll
    for (int r = 0; r < 8; ++r) {
        const int m = r + (half << 3);
        {
            const int bn = m;            // M-tile 0: bins 0..15
            obase[bn] = acc0[r];
        }
        {
            const int bn = 16 + m;       // M-tile 1: bins 16..31
            obase[bn] = acc1[r];
        }
        {
            const int bn = 32 + m;       // M-tile 2: bins 32..47
            obase[bn] = acc2[r];
        }
        {
            const int bn = 48 + m;       // M-tile 3: bin 48 only is valid
            if (bn < OUT_HW * OUT_HW) obase[bn] = acc3[r];
        }
    }
}

extern "C" void kernel_launch(void* const* d_in, const int* in_sizes, int n_in,
                              void* d_out, int out_size, void* d_ws, size_t ws_size,
                              hipStream_t stream) {
    const float* feats = (const float*)d_in[0];
    const float* rois  = (const float*)d_in[1];
    float* out = (float*)d_out;
    (void)in_sizes; (void)n_in; (void)out_size; (void)d_ws; (void)ws_size;

    // 1024 rois x 2 blocks; each 256-thread block = 8 waves,
    // each wave owns one 16-channel tile of one roi.
    roi_align_wmma_kernel<<<dim3(NROI * 2), dim3(256), 0, stream>>>(feats, rois, out);
}